// DFT_75118978007172
// MI455X (gfx1250) — compile-verified
//
#include <hip/hip_runtime.h>

#define N_SAMP   262144
#define N_FFT    2048
#define STRIDE   512
#define N_FRAMES 513
#define HALF     1024

typedef __attribute__((ext_vector_type(2))) float v2f;
typedef __attribute__((ext_vector_type(8))) float v8f;

__device__ __forceinline__ float load_reflect(const float* __restrict__ x, int j) {
  if (j < 0) j = -j;
  if (j >= N_SAMP) j = 2 * N_SAMP - 2 - j;
  return x[j];
}

#define WMMA_F32(a, b, c) \
  __builtin_amdgcn_wmma_f32_16x16x4_f32(false, (a), false, (b), (short)0, (c), false, false)

__global__ __launch_bounds__(256) void stft_wmma_kernel(
    const float* __restrict__ x,
    const float* __restrict__ wsin,
    const float* __restrict__ wcos,
    float* __restrict__ out)
{
  constexpr int KC = 64;          // K-chunk staged in LDS
  constexpr int TT = 32;          // frames (t) per workgroup
  constexpr int PSTRIDE = 66;     // pair-row stride: 2*TT + 2 pad floats
  // Pair-interleaved B tile: Bs[(kk>>1)*PSTRIDE + 2*t + (kk&1)]
  __shared__ float Bs[(KC / 2) * PSTRIDE];

  const int tid  = threadIdx.x;
  const int lane = tid & 31;      // wave32
  const int wave = tid >> 5;      // 0..7
  const int k0 = blockIdx.x * 128;
  const int t0 = blockIdx.y * TT;
  const int b  = blockIdx.z;

  const float* xb = x + (size_t)b * N_SAMP;

  // --- B-fill mapping: thread -> (t, 8 consecutive kk), coalesced 32B reads
  const int ft = tid >> 3;        // t within tile: 0..31
  const int fk = (tid & 7) * 8;   // kk base: 0..56
  int tg = t0 + ft;
  if (tg > N_FRAMES - 1) tg = N_FRAMES - 1;   // clamp overhang (EXEC stays all-1s)
  const int fill_base = tg * STRIDE + fk - HALF;
  float* fill_dst = &Bs[(fk >> 1) * PSTRIDE + 2 * ft];

  // --- A operand per-lane pointers (16x4 fp32 layout: lanes 0-15 K={0,1}, 16-31 K={2,3})
  const int m    = lane & 15;
  const int koff = (lane >> 4) * 2;
  const float* As = wsin + (size_t)(k0 + wave * 16 + m) * N_FFT + koff;
  const float* Ac = wcos + (size_t)(k0 + wave * 16 + m) * N_FFT + koff;

  const int tb = lane & 15;       // B/D column for this lane
  // Per-lane base into the pair-interleaved tile: row = 4s + koff -> pairIdx = 2s + koff/2
  const float* Bbase = &Bs[(koff >> 1) * PSTRIDE + 2 * tb];

  v8f cc0 = {}; v8f cc1 = {};     // cos accumulators (t-tile 0 / 1)
  v8f cs0 = {}; v8f cs1 = {};     // sin accumulators

  for (int c = 0; c < N_FFT / KC; ++c) {
    __syncthreads();
    // Stage frames chunk: B[kk][t] = xp[b][(t0+t)*512 + c*64 + kk], pair-interleaved
    {
      const int j0 = fill_base + c * KC;
      float tmp[8];
      if (j0 >= 0 && j0 + 7 < N_SAMP) {
        const float4 v0 = *(const float4*)(xb + j0);
        const float4 v1 = *(const float4*)(xb + j0 + 4);
        tmp[0] = v0.x; tmp[1] = v0.y; tmp[2] = v0.z; tmp[3] = v0.w;
        tmp[4] = v1.x; tmp[5] = v1.y; tmp[6] = v1.z; tmp[7] = v1.w;
      } else {
        #pragma unroll
        for (int q = 0; q < 8; ++q) tmp[q] = load_reflect(xb, j0 + q);
      }
      #pragma unroll
      for (int q = 0; q < 4; ++q) {
        v2f w; w.x = tmp[2 * q]; w.y = tmp[2 * q + 1];
        *(v2f*)(fill_dst + q * PSTRIDE) = w;   // 8B-aligned ds_store_b64
      }
    }
    __syncthreads();

    if (c + 1 < N_FFT / KC) {           // warm caches (WGP scope) for next chunk's A rows
      __builtin_prefetch(As + (c + 1) * KC, 0, 3);
      __builtin_prefetch(Ac + (c + 1) * KC, 0, 3);
    }

    #pragma unroll
    for (int s = 0; s < KC / 4; ++s) {
      const v2f a_s = *(const v2f*)(As + c * KC + s * 4);
      const v2f a_c = *(const v2f*)(Ac + c * KC + s * 4);
      // B operands: rows (4s+koff, 4s+koff+1) for column tb (tile0) and tb+16 (tile1)
      const float* bp = Bbase + (2 * s) * PSTRIDE;
      const v2f b0 = *(const v2f*)(bp);        // fuses into ds_load_2addr_b64
      const v2f b1 = *(const v2f*)(bp + 32);
      cc0 = WMMA_F32(a_c, b0, cc0);
      cc1 = WMMA_F32(a_c, b1, cc1);
      cs0 = WMMA_F32(a_s, b0, cs0);
      cs1 = WMMA_F32(a_s, b1, cs1);
    }
  }

  // --- Epilogue: D layout VGPR r -> row r + 8*(lane>=16), column lane&15
  const size_t IM = (size_t)8 * N_FFT * N_FRAMES;  // offset of second output (-imag)
  const int kbase = k0 + wave * 16 + 8 * (lane >> 4);
  const int ta  = t0 + tb;
  const int ta2 = ta + 16;
  float* realp = out + (size_t)b * N_FFT * N_FRAMES;
  float* imagp = realp + IM;
  #pragma unroll
  for (int r = 0; r < 8; ++r) {
    const int k = kbase + r;
    if (ta < N_FRAMES) {
      realp[(size_t)k * N_FRAMES + ta] = cc0[r];
      imagp[(size_t)k * N_FRAMES + ta] = -cs0[r];
    }
    if (ta2 < N_FRAMES) {
      realp[(size_t)k * N_FRAMES + ta2] = cc1[r];
      imagp[(size_t)k * N_FRAMES + ta2] = -cs1[r];
    }
  }
}

extern "C" void kernel_launch(void* const* d_in, const int* in_sizes, int n_in,
                              void* d_out, int out_size, void* d_ws, size_t ws_size,
                              hipStream_t stream) {
  const float* x    = (const float*)d_in[0];
  const float* wsin = (const float*)d_in[1];
  const float* wcos = (const float*)d_in[2];
  float* out = (float*)d_out;

  dim3 grid(N_FFT / 128, (N_FRAMES + 31) / 32, 8);  // 16 x 17 x 8 workgroups
  stft_wmma_kernel<<<grid, dim3(256), 0, stream>>>(x, wsin, wcos, out);
}